// GraphConvolution_14250701488867
// MI455X (gfx1250) — compile-verified
//
#include <hip/hip_runtime.h>

typedef __attribute__((ext_vector_type(2))) float v2f;
typedef __attribute__((ext_vector_type(8))) float v8f;

#define GCN_N_NODES 100000
#define GCN_N_EDGES 1600000
#define GCN_D_IN    256
#define GCN_D_OUT   128

// ---------------------------------------------------------------------------
// Kernel 1: h = x @ W + bias   (fp32 WMMA, wave32)
// Grid: 6250 blocks (one per 16-row stripe of x), 256 threads = 8 waves.
// Wave w computes the 16x16 tile (rows m0..m0+15, cols 16w..16w+15),
// accumulating over K=256 in steps of 4 with V_WMMA_F32_16X16X4_F32.
// ---------------------------------------------------------------------------
__global__ __launch_bounds__(256) void gcn_gemm_bias(
    const float* __restrict__ x, const float* __restrict__ W,
    const float* __restrict__ bias, float* __restrict__ h) {
  const int lane = threadIdx.x & 31;
  const int wave = threadIdx.x >> 5;          // 0..7 -> N tile
  const int m0   = blockIdx.x * 16;
  const int n0   = wave * 16;
  const int half = lane >> 4;                 // 0 = lanes 0-15, 1 = lanes 16-31
  const int l15  = lane & 15;

  // A fragment rows: this lane always supplies row (m0 + l15)
  const float* __restrict__ xr = x + (long)(m0 + l15) * GCN_D_IN;

  v8f c = {};
  #pragma unroll 4
  for (int k0 = 0; k0 < GCN_D_IN; k0 += 4) {
    const int ka = k0 + 2 * half;             // K index of VGPR0 for this lane
    // A 16x4: a[v] = x[m][ka + v]  (contiguous -> b64 load)
    v2f a;
    a.x = xr[ka];
    a.y = xr[ka + 1];
    // B 4x16: b[v] = W[(ka + v)][n0 + l15]
    v2f b;
    b.x = W[(long)ka       * GCN_D_OUT + n0 + l15];
    b.y = W[(long)(ka + 1) * GCN_D_OUT + n0 + l15];
    c = __builtin_amdgcn_wmma_f32_16x16x4_f32(
        /*neg_a=*/false, a, /*neg_b=*/false, b,
        /*c_mod=*/(short)0, c, /*reuse_a=*/false, /*reuse_b=*/false);
  }

  // C/D layout: element r lives at (M = m0 + r + 8*half, N = n0 + l15)
  const float bn = bias[n0 + l15];
  #pragma unroll
  for (int r = 0; r < 8; ++r) {
    h[(long)(m0 + r + 8 * half) * GCN_D_OUT + (n0 + l15)] = c[r] + bn;
  }
}

// ---------------------------------------------------------------------------
// Kernel 2: zero the accumulator (float4 grid-stride)
// ---------------------------------------------------------------------------
__global__ __launch_bounds__(256) void gcn_zero_f4(float4* __restrict__ p, long n4) {
  long i = (long)blockIdx.x * blockDim.x + threadIdx.x;
  long stride = (long)gridDim.x * blockDim.x;
  float4 z = make_float4(0.f, 0.f, 0.f, 0.f);
  for (; i < n4; i += stride) p[i] = z;
}

// ---------------------------------------------------------------------------
// Kernel 3: edge scatter  out[row[e]] += val[e] * h[col[e]]
// One wave per edge (grid-stride over waves). Each lane handles 4 consecutive
// output features: one b128 gather from h, then 4 non-returning
// GLOBAL_ATOMIC_ADD_F32 (relaxed, agent scope). h and out both fit in L2.
// ---------------------------------------------------------------------------
__global__ __launch_bounds__(256) void gcn_scatter(
    const int* __restrict__ erow, const int* __restrict__ ecol,
    const float* __restrict__ eval, const float* __restrict__ h,
    float* __restrict__ out, int n_edges) {
  const int lane   = threadIdx.x & 31;
  const int waveId = (int)((blockIdx.x * blockDim.x + threadIdx.x) >> 5);
  const int nWaves = (int)((gridDim.x * blockDim.x) >> 5);

  for (int e = waveId; e < n_edges; e += nWaves) {
    const int   r = erow[e];
    const int   s = ecol[e];
    const float v = eval[e];

    const float4* hr = (const float4*)(h + (long)s * GCN_D_OUT);
    float4 m = hr[lane];                       // 32 lanes * float4 = 128 floats

    float* o = out + (long)r * GCN_D_OUT + lane * 4;
    __hip_atomic_fetch_add(&o[0], v * m.x, __ATOMIC_RELAXED, __HIP_MEMORY_SCOPE_AGENT);
    __hip_atomic_fetch_add(&o[1], v * m.y, __ATOMIC_RELAXED, __HIP_MEMORY_SCOPE_AGENT);
    __hip_atomic_fetch_add(&o[2], v * m.z, __ATOMIC_RELAXED, __HIP_MEMORY_SCOPE_AGENT);
    __hip_atomic_fetch_add(&o[3], v * m.w, __ATOMIC_RELAXED, __HIP_MEMORY_SCOPE_AGENT);
  }
}

// ---------------------------------------------------------------------------
// Kernel 4: in-place ReLU (float4 grid-stride)
// ---------------------------------------------------------------------------
__global__ __launch_bounds__(256) void gcn_relu_f4(float4* __restrict__ p, long n4) {
  long i = (long)blockIdx.x * blockDim.x + threadIdx.x;
  long stride = (long)gridDim.x * blockDim.x;
  for (; i < n4; i += stride) {
    float4 v = p[i];
    v.x = fmaxf(v.x, 0.f);
    v.y = fmaxf(v.y, 0.f);
    v.z = fmaxf(v.z, 0.f);
    v.w = fmaxf(v.w, 0.f);
    p[i] = v;
  }
}

extern "C" void kernel_launch(void* const* d_in, const int* in_sizes, int n_in,
                              void* d_out, int out_size, void* d_ws, size_t ws_size,
                              hipStream_t stream) {
  const float* x    = (const float*)d_in[0];
  const float* W    = (const float*)d_in[1];
  const float* bias = (const float*)d_in[2];
  const int*   erow = (const int*)d_in[3];
  const int*   ecol = (const int*)d_in[4];
  const float* ev   = (const float*)d_in[5];
  float* out = (float*)d_out;
  float* h   = (float*)d_ws;   // 100000*128*4 = 51.2 MB scratch for h = xW+b

  // 1) dense projection via WMMA
  gcn_gemm_bias<<<GCN_N_NODES / 16, 256, 0, stream>>>(x, W, bias, h);

  // 2) zero accumulator (re-done every call -> deterministic under replay)
  const long n4 = (long)GCN_N_NODES * GCN_D_OUT / 4;
  gcn_zero_f4<<<2048, 256, 0, stream>>>((float4*)out, n4);

  // 3) edge scatter with hardware float atomics (L2-resident)
  gcn_scatter<<<8192, 256, 0, stream>>>(erow, ecol, ev, h, out, GCN_N_EDGES);

  // 4) ReLU in place
  gcn_relu_f4<<<2048, 256, 0, stream>>>((float4*)out, n4);
}